// Texture_27212912787616
// MI455X (gfx1250) — compile-verified
//
#include <hip/hip_runtime.h>
#include <hip/hip_bf16.h>

// Multi-level bilinear grid_sample gather, tuned for MI455X (gfx1250, wave32).
// Memory-bound: ~220 MB HBM traffic -> ~9.5us floor at 23.3 TB/s.
// Textures (22 MB total) are kept resident in the 192 MB L2 by streaming the
// coordinate reads and result writes with non-temporal hints (TH=NT), so the
// ~200 MB of stream traffic does not evict the gather working set.
// Gathers are spatially random (grid from uniform-random input) -> no LDS/TDM
// tiling is possible; WMMA has no matmul structure to exploit here.

typedef float f32x4 __attribute__((ext_vector_type(4)));
typedef float f32x2 __attribute__((ext_vector_type(2)));

__device__ __forceinline__ float sample_bilinear(const float* __restrict__ tex,
                                                 int S, float fx, float fy) {
    // ix = ((fx*2-1+1)*S - 1)*0.5 == fx*S - 0.5  (same for iy)
    float ix = fmaf(fx, (float)S, -0.5f);
    float iy = fmaf(fy, (float)S, -0.5f);
    float x0f = floorf(ix);
    float y0f = floorf(iy);
    float wx1 = ix - x0f;
    float wy1 = iy - y0f;
    float wx0 = 1.0f - wx1;
    float wy0 = 1.0f - wy1;
    int x0 = (int)x0f;
    int y0 = (int)y0f;
    int x1 = x0 + 1;
    int y1 = y0 + 1;
    // zeros padding: fold validity into the weights (branchless)
    if (x0 < 0 || x0 >= S) wx0 = 0.0f;
    if (x1 < 0 || x1 >= S) wx1 = 0.0f;
    if (y0 < 0 || y0 >= S) wy0 = 0.0f;
    if (y1 < 0 || y1 >= S) wy1 = 0.0f;
    int xc0 = min(max(x0, 0), S - 1);
    int xc1 = min(max(x1, 0), S - 1);
    int yc0 = min(max(y0, 0), S - 1);
    int yc1 = min(max(y1, 0), S - 1);
    const float* r0 = tex + (size_t)yc0 * (size_t)S;
    const float* r1 = tex + (size_t)yc1 * (size_t)S;
    float t00 = r0[xc0];
    float t01 = r0[xc1];
    float t10 = r1[xc0];
    float t11 = r1[xc1];
    return wy0 * fmaf(wx0, t00, wx1 * t01) + wy1 * fmaf(wx0, t10, wx1 * t11);
}

__device__ __forceinline__ float sample_all(const float* __restrict__ l1,
                                            const float* __restrict__ l2,
                                            const float* __restrict__ l3,
                                            const float* __restrict__ l4,
                                            float fx, float fy) {
    float a = sample_bilinear(l1, 2048, fx, fy);
    float b = sample_bilinear(l2, 1024, fx, fy);
    float c = sample_bilinear(l3, 512,  fx, fy);
    float d = sample_bilinear(l4, 256,  fx, fy);
    return (a + b) + (c + d);
}

__global__ void __launch_bounds__(256)
texture_sample_kernel(const f32x4* __restrict__ x4,   // 2 pixels (4 floats) per elem
                      const float* __restrict__ l1,
                      const float* __restrict__ l2,
                      const float* __restrict__ l3,
                      const float* __restrict__ l4,
                      f32x2* __restrict__ out,         // 2 outputs per elem
                      int npairs) {
    int t = blockIdx.x * blockDim.x + threadIdx.x;
    if (t >= npairs) return;

    // Streaming prefetch of the coordinate stream (gfx1250 global_prefetch_b8,
    // SYS scope -> GL2 only). Speculative: OOB translations silently dropped.
    __builtin_prefetch((const void*)(x4 + t + 4096), 0, 0);

    // Non-temporal load: don't let the 134 MB coordinate stream evict the
    // L2-resident textures.
    f32x4 g = __builtin_nontemporal_load(x4 + t);  // pixel0=(g.x,g.y) pixel1=(g.z,g.w)

    float r0 = sample_all(l1, l2, l3, l4, g.x, g.y);
    float r1 = sample_all(l1, l2, l3, l4, g.z, g.w);

    f32x2 res;
    res.x = r0;
    res.y = r1;
    // Non-temporal store: the 67 MB result stream is write-once.
    __builtin_nontemporal_store(res, out + t);
}

extern "C" void kernel_launch(void* const* d_in, const int* in_sizes, int n_in,
                              void* d_out, int out_size, void* d_ws, size_t ws_size,
                              hipStream_t stream) {
    (void)in_sizes; (void)n_in; (void)d_ws; (void)ws_size;
    const f32x4* x4 = (const f32x4*)d_in[0];
    const float* l1 = (const float*)d_in[1];
    const float* l2 = (const float*)d_in[2];
    const float* l3 = (const float*)d_in[3];
    const float* l4 = (const float*)d_in[4];
    f32x2* out = (f32x2*)d_out;

    int npairs = out_size >> 1;              // 2 pixels per thread
    int threads = 256;                       // 8 wave32 waves per block
    int blocks = (npairs + threads - 1) / threads;
    texture_sample_kernel<<<blocks, threads, 0, stream>>>(x4, l1, l2, l3, l4, out, npairs);
}